// LightGCNLayer_58394375356508
// MI455X (gfx1250) — compile-verified
//
#include <hip/hip_runtime.h>
#include <stdint.h>

#define D_FEAT 64
#define TILE   256   // edges staged per block

// ---------------------------------------------------------------------------
// HOT KERNEL (placed first so the disasm snippet shows it):
// gather-scale-scatter. One 256-edge tile per block.
//  - Edge indices staged into LDS with CDNA5 async global->LDS copies
//    (ASYNCcnt path), released with s_wait_asynccnt + barrier.
//  - 16-lane groups own one edge each: one global_load_b128 of the source
//    row chunk, 4 non-returning native f32 atomics (inline asm,
//    scope:SCOPE_DEV) into the destination row; a group writes 64
//    consecutive dwords -> 2 cache lines.
//  - Next iteration's source row is prefetched (global_prefetch_b8).
// ---------------------------------------------------------------------------
__global__ __launch_bounds__(256) void lgcn_scatter(const float* __restrict__ h,
                                                    const int* __restrict__ esrc,
                                                    const int* __restrict__ edst,
                                                    const float* __restrict__ inv_out,
                                                    const float* __restrict__ inv_in,
                                                    float* __restrict__ out, int E) {
    __shared__ int s_idx[2 * TILE];   // [0,TILE): src   [TILE,2*TILE): dst

    const int tid  = threadIdx.x;
    const int base = blockIdx.x * TILE;

    // ---- async stage of this tile's edge indices into LDS ----
    {
        int e = base + tid;
        if (e < E) {
            // low 32 bits of a generic pointer to LDS == wave-relative LDS byte offset
            unsigned l0 = (unsigned)(uintptr_t)&s_idx[tid];
            unsigned l1 = (unsigned)(uintptr_t)&s_idx[TILE + tid];
            unsigned long long g0 = (unsigned long long)(uintptr_t)(esrc + e);
            unsigned long long g1 = (unsigned long long)(uintptr_t)(edst + e);
            asm volatile("global_load_async_to_lds_b32 %0, %1, off" :: "v"(l0), "v"(g0) : "memory");
            asm volatile("global_load_async_to_lds_b32 %0, %1, off" :: "v"(l1), "v"(g1) : "memory");
        }
        asm volatile("s_wait_asynccnt 0x0" ::: "memory");
    }
    __syncthreads();

    const int wg   = tid >> 4;   // 16 groups of 16 lanes
    const int lane = tid & 15;   // lane covers features [lane*4, lane*4+4)

    for (int it = 0; it < TILE / 16; ++it) {
        const int le = (it << 4) + wg;
        const int e  = base + le;
        if (e < E) {
            const int s = s_idx[le];
            const int d = s_idx[TILE + le];

            // prefetch next iteration's source row chunk (indices already in LDS)
            const int le2 = le + 16;
            if (le2 < TILE && base + le2 < E) {
                __builtin_prefetch(h + (size_t)s_idx[le2] * D_FEAT + lane * 4, 0, 0);
            }

            const float w = inv_out[s] * inv_in[d];
            const float4 v = ((const float4*)(h + (size_t)s * D_FEAT))[lane];

            const float ax = v.x * w, ay = v.y * w, az = v.z * w, aw = v.w * w;
            unsigned long long oaddr =
                (unsigned long long)(uintptr_t)(out + (size_t)d * D_FEAT + lane * 4);
            // native non-returning f32 atomic adds at device scope (L2 atomic units)
            asm volatile("global_atomic_add_f32 %0, %1, off scope:SCOPE_DEV"
                         :: "v"(oaddr), "v"(ax) : "memory");
            asm volatile("global_atomic_add_f32 %0, %1, off offset:4 scope:SCOPE_DEV"
                         :: "v"(oaddr), "v"(ay) : "memory");
            asm volatile("global_atomic_add_f32 %0, %1, off offset:8 scope:SCOPE_DEV"
                         :: "v"(oaddr), "v"(az) : "memory");
            asm volatile("global_atomic_add_f32 %0, %1, off offset:12 scope:SCOPE_DEV"
                         :: "v"(oaddr), "v"(aw) : "memory");
        }
    }
}

// ---------------------------------------------------------------------------
// Zero the output and the degree workspace (must run every launch: graph
// replays do not re-poison, and we accumulate with atomics).
// ---------------------------------------------------------------------------
__global__ __launch_bounds__(256) void lgcn_zero(float4* __restrict__ out4, long long n4,
                                                 uint32_t* __restrict__ deg, int ndeg) {
    long long i      = (long long)blockIdx.x * blockDim.x + threadIdx.x;
    long long stride = (long long)gridDim.x * blockDim.x;
    for (long long k = i; k < n4; k += stride)
        out4[k] = make_float4(0.f, 0.f, 0.f, 0.f);
    for (long long k = i; k < ndeg; k += stride)
        deg[k] = 0u;
}

// ---------------------------------------------------------------------------
// Out/in degree histograms (non-returning u32 atomics, agent scope).
// ---------------------------------------------------------------------------
__global__ __launch_bounds__(256) void lgcn_degree(const int* __restrict__ esrc,
                                                   const int* __restrict__ edst,
                                                   uint32_t* __restrict__ deg_out,
                                                   uint32_t* __restrict__ deg_in, int E) {
    int e = blockIdx.x * blockDim.x + threadIdx.x;
    if (e < E) {
        (void)__hip_atomic_fetch_add(&deg_out[esrc[e]], 1u, __ATOMIC_RELAXED, __HIP_MEMORY_SCOPE_AGENT);
        (void)__hip_atomic_fetch_add(&deg_in [edst[e]], 1u, __ATOMIC_RELAXED, __HIP_MEMORY_SCOPE_AGENT);
    }
}

// ---------------------------------------------------------------------------
// deg -> rsqrt(max(deg,1)), in place (u32 view -> f32 view).
// ---------------------------------------------------------------------------
__global__ __launch_bounds__(256) void lgcn_rsqrt(uint32_t* __restrict__ deg, int n) {
    int i = blockIdx.x * blockDim.x + threadIdx.x;
    if (i < n) {
        float d = (float)deg[i];
        ((float*)deg)[i] = rsqrtf(fmaxf(d, 1.0f));
    }
}

// ---------------------------------------------------------------------------
// Launch: inputs are h [N*64 f32], edge_src [E i32], edge_dst [E i32];
// output is h_out [N*64 f32]. Workspace: 2*N u32 degree/inv-sqrt arrays.
// ---------------------------------------------------------------------------
extern "C" void kernel_launch(void* const* d_in, const int* in_sizes, int n_in,
                              void* d_out, int out_size, void* d_ws, size_t ws_size,
                              hipStream_t stream) {
    const float* h    = (const float*)d_in[0];
    const int*   esrc = (const int*)d_in[1];
    const int*   edst = (const int*)d_in[2];
    float*       out  = (float*)d_out;

    const int N = in_sizes[0] / D_FEAT;
    const int E = in_sizes[1];

    uint32_t* deg_out = (uint32_t*)d_ws;       // N u32, becomes f32 inv_sqrt_out
    uint32_t* deg_in  = deg_out + N;           // N u32, becomes f32 inv_sqrt_in
    (void)ws_size;                             // requires 2*N*4 = 800 KB

    const long long n4 = (long long)N * (D_FEAT / 4);
    lgcn_zero<<<2048, 256, 0, stream>>>((float4*)out, n4, deg_out, 2 * N);

    const int eblocks = (E + 255) / 256;
    lgcn_degree<<<eblocks, 256, 0, stream>>>(esrc, edst, deg_out, deg_in, E);

    lgcn_rsqrt<<<(2 * N + 255) / 256, 256, 0, stream>>>(deg_out, 2 * N);

    const int sblocks = (E + TILE - 1) / TILE;
    lgcn_scatter<<<sblocks, 256, 0, stream>>>(h, esrc, edst,
                                              (const float*)deg_out, (const float*)deg_in,
                                              out, E);
}